// CLAM_Hierarchical_69518340653574
// MI455X (gfx1250) — compile-verified
//
#include <hip/hip_runtime.h>
#include <math.h>

// ---------------------------------------------------------------------------
// CDNA5 (gfx1250) implementation of CLAM hierarchical attention.
// f32 WMMA (V_WMMA_F32_16X16X4_F32): workload is memory-bound (fp32
// activations ~184MB @ 23.3TB/s >> f32-WMMA compute time), so stay in f32.
// 32 rows/block with register B-reuse halves L2 weight traffic; h tiles are
// staged to LDS once via async global->LDS (ASYNCcnt) instead of 8x redundant
// per-wave global reads.
// ---------------------------------------------------------------------------

typedef float v2f __attribute__((ext_vector_type(2)));
typedef float v8f __attribute__((ext_vector_type(8)));

static __device__ __forceinline__ v8f wmma4(v2f a, v2f b, v8f c) {
  // D = A(16x4 f32) * B(4x16 f32) + C(16x16 f32)
  return __builtin_amdgcn_wmma_f32_16x16x4_f32(false, a, false, b, (short)0, c,
                                               false, false);
}

static __device__ __forceinline__ v8f v8zero() {
  v8f r;
#pragma unroll
  for (int i = 0; i < 8; ++i) r[i] = 0.f;
  return r;
}

static __device__ __forceinline__ unsigned mapkey(float f) {
  unsigned u = __float_as_uint(f);
  return (u & 0x80000000u) ? ~u : (u | 0x80000000u);
}

#define OFFC 84.0f /* 224/2 - 224/4/2 */
#define WINC 225.0f
#define HSTR 1028  /* padded f32 stride for 1024-col h rows (bank = 4m+k) */
#define XSTR 516   /* padded f32 stride for 512-col x rows */

// Stage a 32-row x 1024-col f32 tile into LDS with async global->LDS b128.
// Each chunk = one wave-instruction = 32 lanes x 16B = 512B. 256 chunks.
static __device__ __forceinline__ void stage_h_async(
    const float* __restrict__ H, const int* sidx, float* shh, int wv, int l) {
  unsigned ldsBase = (unsigned)(size_t)shh;
  for (int c = wv; c < 256; c += 8) {
    int row = c >> 3;      // 8 chunks of 512B per 4KB row
    int sub = c & 7;
    unsigned ldsa =
        ldsBase + (unsigned)row * (HSTR * 4u) + (unsigned)sub * 512u +
        (unsigned)(l << 4);
    const char* g = (const char*)H + ((size_t)sidx[row] << 12) +
                    (size_t)((sub << 9) + (l << 4));
    asm volatile("global_load_async_to_lds_b128 %0, %1, off" ::"v"(ldsa),
                 "v"(g)
                 : "memory");
  }
  asm volatile("s_wait_asynccnt 0x0" ::: "memory");
}

// ---------------------------------------------------------------------------
// Fused attention-score kernel: 32 selected rows per block (2 row-tiles).
//   x = relu(h@W1+b1)  (K=1024 -> 512, LDS-resident)
//   A = (tanh(x@Wa+ba)*sigmoid(x@Wb+bb))@Wc + bc
// 8 waves. Stage1: wave owns 64 cols, B-frags reused across both row-tiles.
// Stage2: wave owns 32 cols of a and g. Deterministic slot reductions.
// Dynamic LDS: h 32*1028 + x 32*516 + sc 32*128 floats (~214KB).
// ---------------------------------------------------------------------------
__global__ __launch_bounds__(256) void attnA_kernel(
    const float* __restrict__ H, const int* __restrict__ idx,
    const int* __restrict__ cntPtr, int fixedCnt,
    const float* __restrict__ W1, const float* __restrict__ Wa,
    const float* __restrict__ Wb, const float* __restrict__ Wc,
    const float* __restrict__ b1, const float* __restrict__ ba,
    const float* __restrict__ bb, const float* __restrict__ bc,
    float* __restrict__ Aout) {
  extern __shared__ float smem[];
  float* shh = smem;                       // [32][HSTR]
  float* sx = smem + 32 * HSTR;            // [32][XSTR]
  float* sc = smem + 32 * HSTR + 32 * XSTR;  // [32][128]
  __shared__ int sidx[32];

  int cnt = cntPtr ? *cntPtr : fixedCnt;
  int rowbase = blockIdx.x * 32;
  if (rowbase >= cnt) return;  // uniform (before any barrier)

  int tid = threadIdx.x;
  if (tid < 32) {
    int g = rowbase + tid;
    sidx[tid] = (g < cnt) ? (idx ? idx[g] : g) : 0;
  }
  __syncthreads();

  int wv = tid >> 5;  // wave 0..7
  int l = tid & 31;
  int m = l & 15;
  int kh = l >> 4;

  stage_h_async(H, sidx, shh, wv, l);
  __syncthreads();

  // ---------------- stage 1: x = relu(h @ W1 + b1) ----------------
  v8f acc[2][4];
#pragma unroll
  for (int rt = 0; rt < 2; ++rt)
#pragma unroll
    for (int t = 0; t < 4; ++t) acc[rt][t] = v8zero();

  for (int k = 0; k < 1024; k += 4) {
    int kk = k + 2 * kh;
    v2f a0 = *(const v2f*)(shh + m * HSTR + kk);
    v2f a1 = *(const v2f*)(shh + (16 + m) * HSTR + kk);
#pragma unroll
    for (int t = 0; t < 4; ++t) {
      int col = (wv << 6) + (t << 4) + m;
      v2f b = {W1[kk * 512 + col], W1[(kk + 1) * 512 + col]};
      acc[0][t] = wmma4(a0, b, acc[0][t]);
      acc[1][t] = wmma4(a1, b, acc[1][t]);
    }
  }
#pragma unroll
  for (int t = 0; t < 4; ++t) {
    int col = (wv << 6) + (t << 4) + m;
    float bias = b1[col];
#pragma unroll
    for (int rt = 0; rt < 2; ++rt) {
#pragma unroll
      for (int r = 0; r < 8; ++r) {
        float v = acc[rt][t][r] + bias;
        sx[(rt * 16 + r + 8 * kh) * XSTR + col] = v > 0.f ? v : 0.f;
      }
    }
  }
  __syncthreads();

  // ---------------- stage 2: a,g tiles + gated dot with Wc ----------------
  v8f aA[2][2], aG[2][2];
#pragma unroll
  for (int rt = 0; rt < 2; ++rt)
#pragma unroll
    for (int t = 0; t < 2; ++t) { aA[rt][t] = v8zero(); aG[rt][t] = v8zero(); }

  int c0 = (wv << 5) + m, c1 = c0 + 16;
  for (int k = 0; k < 512; k += 4) {
    int kk = k + 2 * kh;
    v2f x0 = *(const v2f*)(sx + m * XSTR + kk);
    v2f x1 = *(const v2f*)(sx + (16 + m) * XSTR + kk);
    v2f bA0 = {Wa[kk * 256 + c0], Wa[(kk + 1) * 256 + c0]};
    v2f bA1 = {Wa[kk * 256 + c1], Wa[(kk + 1) * 256 + c1]};
    v2f bG0 = {Wb[kk * 256 + c0], Wb[(kk + 1) * 256 + c0]};
    v2f bG1 = {Wb[kk * 256 + c1], Wb[(kk + 1) * 256 + c1]};
    aA[0][0] = wmma4(x0, bA0, aA[0][0]);
    aA[0][1] = wmma4(x0, bA1, aA[0][1]);
    aA[1][0] = wmma4(x1, bA0, aA[1][0]);
    aA[1][1] = wmma4(x1, bA1, aA[1][1]);
    aG[0][0] = wmma4(x0, bG0, aG[0][0]);
    aG[0][1] = wmma4(x0, bG1, aG[0][1]);
    aG[1][0] = wmma4(x1, bG0, aG[1][0]);
    aG[1][1] = wmma4(x1, bG1, aG[1][1]);
  }
  float ba0 = ba[c0], ba1 = ba[c1], bb0 = bb[c0], bb1 = bb[c1];
  float wc0 = Wc[c0], wc1 = Wc[c1];
#pragma unroll
  for (int rt = 0; rt < 2; ++rt) {
#pragma unroll
    for (int r = 0; r < 8; ++r) {
      float t0 =
          tanhf(aA[rt][0][r] + ba0) * (1.f / (1.f + expf(-(aG[rt][0][r] + bb0))));
      float t1 =
          tanhf(aA[rt][1][r] + ba1) * (1.f / (1.f + expf(-(aG[rt][1][r] + bb1))));
      int row = rt * 16 + r + 8 * kh;
      sc[row * 128 + (wv << 4) + m] = t0 * wc0 + t1 * wc1;  // unique slot
    }
  }
  __syncthreads();
  if (tid < 32) {
    int g = rowbase + tid;
    if (g < cnt) {
      float s = 0.f;
      const float* p = sc + tid * 128;
      for (int j = 0; j < 128; ++j) s += p[j];
      Aout[g] = s + bc[0];
    }
  }
}

// ---------------------------------------------------------------------------
// Weighted x recompute (final level): partial[tile][col] = sum_m w_m*x[m][col]
// 32 rows/block; deterministic, no atomics.
// ---------------------------------------------------------------------------
__global__ __launch_bounds__(256) void wx_partial_kernel(
    const float* __restrict__ H, const int* __restrict__ idx,
    const int* __restrict__ cntPtr, const float* __restrict__ W1,
    const float* __restrict__ b1, const float* __restrict__ wts,
    float* __restrict__ partial) {
  extern __shared__ float smem[];
  float* shh = smem;  // [32][HSTR]
  __shared__ int sidx[32];
  __shared__ float swt[32];
  __shared__ float spart[512 * 2];
  int tid = threadIdx.x;
  for (int c = tid; c < 512; c += 256) partial[blockIdx.x * 512 + c] = 0.f;
  int cnt = *cntPtr;
  int rowbase = blockIdx.x * 32;
  if (rowbase >= cnt) return;
  if (tid < 32) {
    int g = rowbase + tid;
    if (g < cnt) { sidx[tid] = idx ? idx[g] : g; swt[tid] = wts[g]; }
    else         { sidx[tid] = 0;                swt[tid] = 0.f;    }
  }
  __syncthreads();
  int wv = tid >> 5, l = tid & 31, m = l & 15, kh = l >> 4;

  stage_h_async(H, sidx, shh, wv, l);
  __syncthreads();

  v8f acc[2][4];
#pragma unroll
  for (int rt = 0; rt < 2; ++rt)
#pragma unroll
    for (int t = 0; t < 4; ++t) acc[rt][t] = v8zero();
  for (int k = 0; k < 1024; k += 4) {
    int kk = k + 2 * kh;
    v2f a0 = *(const v2f*)(shh + m * HSTR + kk);
    v2f a1 = *(const v2f*)(shh + (16 + m) * HSTR + kk);
#pragma unroll
    for (int t = 0; t < 4; ++t) {
      int col = (wv << 6) + (t << 4) + m;
      v2f b = {W1[kk * 512 + col], W1[(kk + 1) * 512 + col]};
      acc[0][t] = wmma4(a0, b, acc[0][t]);
      acc[1][t] = wmma4(a1, b, acc[1][t]);
    }
  }
#pragma unroll
  for (int t = 0; t < 4; ++t) {
    int col = (wv << 6) + (t << 4) + m;
    float bias = b1[col];
    float s = 0.f;
#pragma unroll
    for (int rt = 0; rt < 2; ++rt) {
#pragma unroll
      for (int r = 0; r < 8; ++r) {
        float x = acc[rt][t][r] + bias;
        x = x > 0.f ? x : 0.f;
        s += swt[rt * 16 + r + 8 * kh] * x;
      }
    }
    spart[col * 2 + kh] = s;
  }
  __syncthreads();
  for (int c = tid; c < 512; c += 256)
    partial[blockIdx.x * 512 + c] = spart[c * 2] + spart[c * 2 + 1];
}

// ------------------------- top-k / match / compact -------------------------
__global__ void initc_kernel(int* counters) {
  if (threadIdx.x >= 2 && threadIdx.x < 8) counters[threadIdx.x] = 0;
}

// exact k-th largest via 4x8-bit radix descent on order-mapped keys
__global__ __launch_bounds__(1024) void radix_select_kernel(
    const float* __restrict__ A, const int* __restrict__ cntPtr, int fixedCnt,
    int* __restrict__ counters) {
  __shared__ int hist[256];
  __shared__ unsigned sprefix;
  __shared__ int skrem;
  int cnt = cntPtr ? *cntPtr : fixedCnt;
  if (cnt < 1) cnt = 1;
  int k = (int)(0.3f * (float)cnt);
  if (k < 1) k = 1;
  if (k > cnt) k = cnt;
  if (threadIdx.x == 0) { sprefix = 0u; skrem = k; counters[5] = k; }
  __syncthreads();
  for (int pass = 3; pass >= 0; --pass) {
    int shift = pass * 8;
    for (int b = threadIdx.x; b < 256; b += blockDim.x) hist[b] = 0;
    __syncthreads();
    unsigned pref = sprefix;
    unsigned pmask = (pass == 3) ? 0u : (0xFFFFFFFFu << (shift + 8));
    for (int i = threadIdx.x; i < cnt; i += blockDim.x) {
      unsigned key = mapkey(A[i]);
      if ((key & pmask) == pref) atomicAdd(&hist[(key >> shift) & 255], 1);
    }
    __syncthreads();
    if (threadIdx.x == 0) {
      int krem = skrem;
      for (int b = 255; b >= 0; --b) {
        if (krem <= hist[b]) {
          sprefix = pref | ((unsigned)b << shift);
          skrem = krem;
          break;
        }
        krem -= hist[b];
      }
    }
    __syncthreads();
  }
  if (threadIdx.x == 0) counters[6] = (int)sprefix;
}

__global__ void topk_count_kernel(const float* __restrict__ A,
                                  const int* __restrict__ cntPtr, int fixedCnt,
                                  int* __restrict__ counters) {
  int cnt = cntPtr ? *cntPtr : fixedCnt;
  int i = blockIdx.x * 256 + threadIdx.x;
  if (i >= cnt) return;
  if (mapkey(A[i]) > (unsigned)counters[6]) atomicAdd(&counters[3], 1);
}

__global__ void topk_select_kernel(const float* __restrict__ A,
                                   const int* __restrict__ cntPtr, int fixedCnt,
                                   int* __restrict__ counters,
                                   int* __restrict__ sel) {
  int cnt = cntPtr ? *cntPtr : fixedCnt;
  int i = blockIdx.x * 256 + threadIdx.x;
  if (i >= cnt) return;
  unsigned key = mapkey(A[i]);
  unsigned thr = (unsigned)counters[6];
  if (key > thr) {
    sel[atomicAdd(&counters[2], 1)] = i;
  } else if (key == thr) {
    int t = atomicAdd(&counters[4], 1);
    if (counters[3] + t < counters[5]) sel[atomicAdd(&counters[2], 1)] = i;
  }
}

__global__ __launch_bounds__(256) void match_kernel(
    const int* __restrict__ sel, const int* __restrict__ counters,
    const int* __restrict__ curIdx, const float* __restrict__ cSrc,
    const float* __restrict__ cTgt, int Ntgt, float sf,
    int* __restrict__ mask) {
  __shared__ float cx[1024], cy[1024];
  int selCnt = counters[2];
  int j = blockIdx.x * 256 + threadIdx.x;
  float tx = 0.f, ty = 0.f;
  if (j < Ntgt) { tx = cTgt[2 * j]; ty = cTgt[2 * j + 1]; }
  float rad2 = (WINC * sf) * (WINC * sf);
  int hit = 0;
  for (int base = 0; base < selCnt; base += 1024) {
    int nload = selCnt - base;
    if (nload > 1024) nload = 1024;
    for (int t = threadIdx.x; t < nload; t += 256) {
      int p = sel[base + t];
      int orig = curIdx ? curIdx[p] : p;
      cx[t] = (cSrc[2 * orig] + OFFC) * sf;
      cy[t] = (cSrc[2 * orig + 1] + OFFC) * sf;
    }
    __syncthreads();
    if (j < Ntgt && !hit) {
      for (int t = 0; t < nload; ++t) {
        float dx = tx - cx[t], dy = ty - cy[t];
        if (dx * dx + dy * dy < rad2) { hit = 1; break; }
      }
    }
    __syncthreads();
  }
  if (j < Ntgt) mask[j] = hit;
}

// ordered compaction (matches jnp.nonzero ordering), single block
__global__ __launch_bounds__(1024) void compact_kernel(
    const int* __restrict__ mask, int N, int* __restrict__ outIdx,
    int* __restrict__ counters) {
  __shared__ int scan[1024];
  __shared__ int sbase;
  if (threadIdx.x == 0) sbase = 0;
  __syncthreads();
  for (int base = 0; base < N; base += 1024) {
    int i = base + threadIdx.x;
    int v = (i < N) ? mask[i] : 0;
    scan[threadIdx.x] = v;
    __syncthreads();
    for (int off = 1; off < 1024; off <<= 1) {
      int add = (threadIdx.x >= off) ? scan[threadIdx.x - off] : 0;
      __syncthreads();
      scan[threadIdx.x] += add;
      __syncthreads();
    }
    int incl = scan[threadIdx.x];
    if (v) outIdx[sbase + incl - 1] = i;
    __syncthreads();
    if (threadIdx.x == 1023) sbase += incl;
    __syncthreads();
  }
  if (threadIdx.x == 0) { counters[0] = sbase; counters[1] = sbase; }
}

// ------------------------- final head -------------------------
__global__ __launch_bounds__(1024) void softmax_kernel(
    const float* __restrict__ A, const int* __restrict__ cntPtr,
    float* __restrict__ wout) {
  __shared__ float red[1024];
  __shared__ float smax, ssum;
  int cnt = *cntPtr;
  float m = -3.4e38f;
  for (int i = threadIdx.x; i < cnt; i += 1024) m = fmaxf(m, A[i]);
  red[threadIdx.x] = m;
  __syncthreads();
  for (int off = 512; off; off >>= 1) {
    if (threadIdx.x < off)
      red[threadIdx.x] = fmaxf(red[threadIdx.x], red[threadIdx.x + off]);
    __syncthreads();
  }
  if (threadIdx.x == 0) smax = red[0];
  __syncthreads();
  float s = 0.f;
  for (int i = threadIdx.x; i < cnt; i += 1024) s += expf(A[i] - smax);
  red[threadIdx.x] = s;
  __syncthreads();
  for (int off = 512; off; off >>= 1) {
    if (threadIdx.x < off) red[threadIdx.x] += red[threadIdx.x + off];
    __syncthreads();
  }
  if (threadIdx.x == 0) ssum = red[0];
  __syncthreads();
  float inv = 1.f / ssum;
  for (int i = threadIdx.x; i < cnt; i += 1024) wout[i] = expf(A[i] - smax) * inv;
}

__global__ void copyAraw_kernel(const float* __restrict__ A,
                                const int* __restrict__ cntPtr,
                                float* __restrict__ out, int n) {
  int i = blockIdx.x * 256 + threadIdx.x;
  if (i < n) out[i] = (i < *cntPtr) ? A[i] : 0.f;
}

__global__ void wx_reduce_kernel(const float* __restrict__ partial, int ntiles,
                                 float* __restrict__ Mvec) {
  int c = blockIdx.x * 256 + threadIdx.x;
  if (c < 512) {
    float s = 0.f;
    for (int t = 0; t < ntiles; ++t) s += partial[t * 512 + c];
    Mvec[c] = s;
  }
}

__global__ __launch_bounds__(64) void logits_kernel(
    const float* __restrict__ Mvec, const float* __restrict__ Wcls,
    const float* __restrict__ bcls, float* __restrict__ out) {
  __shared__ float p0[64], p1[64];
  float s0 = 0.f, s1 = 0.f;
  for (int n = threadIdx.x; n < 512; n += 64) {
    float v = Mvec[n];
    s0 += v * Wcls[n * 2];
    s1 += v * Wcls[n * 2 + 1];
  }
  p0[threadIdx.x] = s0;
  p1[threadIdx.x] = s1;
  __syncthreads();
  if (threadIdx.x == 0) {
    float l0 = bcls[0], l1 = bcls[1];
    for (int t = 0; t < 64; ++t) { l0 += p0[t]; l1 += p1[t]; }
    out[0] = l0;
    out[1] = l1;
    float mx = fmaxf(l0, l1);
    float e0 = expf(l0 - mx), e1 = expf(l1 - mx), inv = 1.f / (e0 + e1);
    out[2] = e0 * inv;
    out[3] = e1 * inv;
    out[4] = (l1 > l0) ? 1.f : 0.f;
  }
}

// ---------------------------------------------------------------------------
extern "C" void kernel_launch(void* const* d_in, const int* in_sizes, int n_in,
                              void* d_out, int out_size, void* d_ws,
                              size_t ws_size, hipStream_t stream) {
  (void)in_sizes; (void)n_in; (void)ws_size;
  // Input order (setup_inputs dict; params pytree-flattened, sorted keys):
  // 0..3: h0,h1,h2,h3  4..7: coords0..3  8: Wcls  9: bcls
  // 10+8p: level p {W1,Wa,Wb,Wc,b1,ba,bb,bc}
  const float* hs[4] = {(const float*)d_in[3], (const float*)d_in[2],
                        (const float*)d_in[1], (const float*)d_in[0]};
  const float* cs[4] = {(const float*)d_in[7], (const float*)d_in[6],
                        (const float*)d_in[5], (const float*)d_in[4]};
  const int Ns[4] = {1000, 4000, 8000, 32000};
  const float* Wcls = (const float*)d_in[8];
  const float* bcls = (const float*)d_in[9];
  struct Lvl { const float *W1, *Wa, *Wb, *Wc, *b1, *ba, *bb, *bc; } L[4];
  for (int p = 0; p < 4; ++p) {
    int b = 10 + 8 * p;
    L[p].W1 = (const float*)d_in[b + 0]; L[p].Wa = (const float*)d_in[b + 1];
    L[p].Wb = (const float*)d_in[b + 2]; L[p].Wc = (const float*)d_in[b + 3];
    L[p].b1 = (const float*)d_in[b + 4]; L[p].ba = (const float*)d_in[b + 5];
    L[p].bb = (const float*)d_in[b + 6]; L[p].bc = (const float*)d_in[b + 7];
  }
  // workspace carve (~2.8MB)
  char* w = (char*)d_ws;
  auto carve = [&](size_t bytes) {
    char* p = w;
    w += (bytes + 255) & ~(size_t)255;
    return p;
  };
  int* idxA = (int*)carve(32000 * 4);
  int* idxB = (int*)carve(32000 * 4);
  float* Abuf = (float*)carve(32000 * 4);
  float* wbuf = (float*)carve(32000 * 4);
  int* sel = (int*)carve(16384 * 4);
  int* mask = (int*)carve(32000 * 4);
  float* partial = (float*)carve((size_t)1000 * 512 * 4);
  float* Mvec = (float*)carve(512 * 4);
  int* counters = (int*)carve(64 * 4);

  const size_t ldsA = (size_t)(32 * HSTR + 32 * XSTR + 32 * 128) * sizeof(float);
  const size_t ldsW = (size_t)(32 * HSTR) * sizeof(float);
  const float SFv = 4.0f;
  int* cur = nullptr;          // null => identity index list
  const int* cntPtr = nullptr; // null => use fixed count
  int flip = 0;

  for (int i = 0; i < 3; ++i) {
    int maxRows = Ns[i];
    attnA_kernel<<<dim3((maxRows + 31) / 32), dim3(256), ldsA, stream>>>(
        hs[i], cur, cntPtr, maxRows, L[i].W1, L[i].Wa, L[i].Wb, L[i].Wc,
        L[i].b1, L[i].ba, L[i].bb, L[i].bc, Abuf);
    initc_kernel<<<dim3(1), dim3(32), 0, stream>>>(counters);
    radix_select_kernel<<<dim3(1), dim3(1024), 0, stream>>>(Abuf, cntPtr,
                                                            maxRows, counters);
    topk_count_kernel<<<dim3((maxRows + 255) / 256), dim3(256), 0, stream>>>(
        Abuf, cntPtr, maxRows, counters);
    topk_select_kernel<<<dim3((maxRows + 255) / 256), dim3(256), 0, stream>>>(
        Abuf, cntPtr, maxRows, counters, sel);
    int* nxt = flip ? idxB : idxA;
    match_kernel<<<dim3((Ns[i + 1] + 255) / 256), dim3(256), 0, stream>>>(
        sel, counters, cur, cs[i], cs[i + 1], Ns[i + 1], SFv, mask);
    compact_kernel<<<dim3(1), dim3(1024), 0, stream>>>(mask, Ns[i + 1], nxt,
                                                       counters);
    cur = nxt;
    cntPtr = counters;  // counters[0] = surviving count
    flip ^= 1;
  }

  // final level (selected rows of h0)
  attnA_kernel<<<dim3(1000), dim3(256), ldsA, stream>>>(
      hs[3], cur, cntPtr, Ns[3], L[3].W1, L[3].Wa, L[3].Wb, L[3].Wc, L[3].b1,
      L[3].ba, L[3].bb, L[3].bc, Abuf);
  softmax_kernel<<<dim3(1), dim3(1024), 0, stream>>>(Abuf, counters, wbuf);
  int nAraw = out_size - 5;
  if (nAraw < 0) nAraw = 0;
  copyAraw_kernel<<<dim3(((nAraw > 0 ? nAraw : 1) + 255) / 256), dim3(256), 0,
                    stream>>>(Abuf, counters, (float*)d_out + 5, nAraw);
  wx_partial_kernel<<<dim3(1000), dim3(256), ldsW, stream>>>(
      hs[3], cur, counters, L[3].W1, L[3].b1, wbuf, partial);
  wx_reduce_kernel<<<dim3(2), dim3(256), 0, stream>>>(partial, 1000, Mvec);
  logits_kernel<<<dim3(1), dim3(64), 0, stream>>>(Mvec, Wcls, bcls,
                                                  (float*)d_out);
}